// GaussianLayer_60017872995170
// MI455X (gfx1250) — compile-verified
//
#include <hip/hip_runtime.h>

typedef float v2f __attribute__((ext_vector_type(2)));
typedef float v8f __attribute__((ext_vector_type(8)));

#define NSIDE 128
#define VOL   (128*128*128)   // elements per channel
#define KS    21
#define PADR  10
#define BAND  36              // 16 outputs + 20 halo = K extent of banded matmul
#define LDSP  20              // padded LDS row stride: mult-of-4 (B128 align), conflict-free
#define WAVES 4               // waves (tiles) per block

// Recover the 1D Gaussian g[0..20] from the separable 3D kernel:
// k3[t][10][10] = g[t]*g[10]^2, and k3[10][10][10] = g[10]^3.
__global__ void make_g1d(const float* __restrict__ W, float* __restrict__ g) {
  int t = threadIdx.x;
  if (t < KS) {
    float s   = W[10 * 441 + 220];   // g10^3
    float g10 = cbrtf(s);
    g[t] = W[t * 441 + 220] / (g10 * g10);
  }
}

// One wave computes a 16(conv axis) x 16(orthogonal axis) output tile of a
// 21-tap clamp-padded 1D convolution as a banded 16x36 * 36x16 f32 matmul:
// 9 chained V_WMMA_F32_16X16X4_F32. Strides are compile-time so all global
// accesses use immediate offsets; wide (B128) transfers where contiguity allows.
template <bool AXIS_X, int SA, int SU, int SV>
__global__ __launch_bounds__(WAVES * 32)
void gauss1d_wmma(const float* __restrict__ src, float* __restrict__ dst,
                  const float* __restrict__ g1d) {
  __shared__ float tile[WAVES][BAND * LDSP];
  __shared__ float wpad[64];   // wpad[i] = g[i-15] if 0<=i-15<=20 else 0

  const int tid  = threadIdx.x;
  const int lane = tid & 31;
  const int wv   = tid >> 5;

  if (tid < 64) {
    int t = tid - 15;
    wpad[tid] = ((unsigned)t <= 20u) ? g1d[t] : 0.0f;
  }

  const int p0 = blockIdx.x * 16;                    // conv-axis tile base
  const int u0 = (blockIdx.y * WAVES + wv) * 16;     // orthogonal tile base
  const int v  = blockIdx.z & (NSIDE - 1);
  const int ch = blockIdx.z >> 7;

  const size_t base = (size_t)ch * VOL + (size_t)v * (size_t)SV;
  const float* s  = src + base;
  float*       d  = dst + base;
  float*       my = tile[wv];

  // ---- Stage 36x16 input tile into LDS (replication-pad via clamp) ----
  if (AXIS_X) {
    // conv axis is unit-stride; clamp breaks per-lane contiguity -> scalar,
    // lanes walk consecutive x -> coalesced.
    #pragma unroll
    for (int i = 0; i < 18; ++i) {                   // 18*32 = 576 = 36*16
      int idx = i * 32 + lane;
      int k = idx % BAND, u = idx / BAND;
      int a = p0 + k - PADR;
      a = a < 0 ? 0 : (a > NSIDE - 1 ? NSIDE - 1 : a);
      my[k * LDSP + u] = s[(size_t)a * SA + (size_t)(u0 + u) * SU];
    }
  } else {
    // orthogonal axis is unit-stride: one float4 per lane, 4 lanes per row.
    #pragma unroll
    for (int i = 0; i < 5; ++i) {                    // 144 float4 = 576 floats
      int idx = i * 32 + lane;
      if (idx < 144) {
        int k  = idx >> 2;                           // row 0..35
        int uq = (idx & 3) * 4;                      // 0,4,8,12
        int a = p0 + k - PADR;
        a = a < 0 ? 0 : (a > NSIDE - 1 ? NSIDE - 1 : a);
        const float4 val =
            *(const float4*)(s + (size_t)a * SA + (size_t)(u0 + uq));
        *(float4*)(&my[k * LDSP + uq]) = val;        // 16B aligned (LDSP%4==0)
      }
    }
  }
  __syncthreads();

  const int n  = lane & 15;                // A row M / B column N for this lane
  const int k0 = (lane >> 4) << 1;         // K base: lanes 0-15 -> 0, 16-31 -> 2
  const int ai = k0 - n + 15;              // wpad base index (always >= 0)
  const int bi = k0 * LDSP + n;            // tile base index

  // Bulk-load all operands: unconditional ds_loads, then 9 chained WMMAs.
  float areg[18], breg[18];
  #pragma unroll
  for (int q = 0; q < 9; ++q) {
    areg[2 * q]     = wpad[ai + 4 * q];
    areg[2 * q + 1] = wpad[ai + 4 * q + 1];
    breg[2 * q]     = my[bi + (4 * q) * LDSP];
    breg[2 * q + 1] = my[bi + (4 * q + 1) * LDSP];
  }

  v8f c = {};
  #pragma unroll
  for (int q = 0; q < 9; ++q) {
    v2f a, b;
    a.x = areg[2 * q]; a.y = areg[2 * q + 1];
    b.x = breg[2 * q]; b.y = breg[2 * q + 1];
    c = __builtin_amdgcn_wmma_f32_16x16x4_f32(
        /*neg_a=*/false, a, /*neg_b=*/false, b,
        /*c_mod=*/(short)0, c, /*reuse_a=*/false, /*reuse_b=*/false);
  }

  // D layout: VGPR r holds D[r + (lane<16?0:8)][lane%16]
  const int mhi = (lane >> 4) << 3;

  if (AXIS_X) {
    // Conv axis unit-stride: transpose through LDS, then 2x B128 store/lane.
    __syncthreads();
    #pragma unroll
    for (int r = 0; r < 8; ++r)
      my[n * LDSP + mhi + r] = c[r];       // store as [u][p]
    __syncthreads();
    #pragma unroll
    for (int i = 0; i < 2; ++i) {          // 64 float4 = 256 outputs
      int idx = i * 32 + lane;
      int u = idx >> 2, p4 = (idx & 3) * 4;
      const float4 val = *(const float4*)(&my[u * LDSP + p4]);
      *(float4*)(d + (size_t)(u0 + u) * SU + (size_t)(p0 + p4)) = val;
    }
  } else {
    // Orthogonal axis unit-stride: lanes 0-15 contiguous; constant-offset clause.
    float* dp = d + (size_t)(u0 + n) * SU + (size_t)(p0 + mhi) * SA;
    #pragma unroll
    for (int r = 0; r < 8; ++r)
      dp[(size_t)r * SA] = c[r];
  }
}

extern "C" void kernel_launch(void* const* d_in, const int* in_sizes, int n_in,
                              void* d_out, int out_size, void* d_ws, size_t ws_size,
                              hipStream_t stream) {
  const float* x = (const float*)d_in[0];
  const float* W = (const float*)d_in[1];
  float* out = (float*)d_out;
  float* g   = (float*)d_ws;                         // 21 floats
  float* tmp = (float*)((char*)d_ws + 256);          // 3*128^3 floats scratch

  make_g1d<<<dim3(1), dim3(32), 0, stream>>>(W, g);

  dim3 grid(NSIDE / 16, (NSIDE / 16) / WAVES, 3 * NSIDE);  // (8, 2, 384)
  dim3 blk(WAVES * 32);

  // Pass 1: conv along z, tile x (unit stride), slice y:  x -> out
  gauss1d_wmma<false, 16384, 1, 128><<<grid, blk, 0, stream>>>(x, out, g);
  // Pass 2: conv along y, tile x (unit stride), slice z:  out -> tmp
  gauss1d_wmma<false, 128, 1, 16384><<<grid, blk, 0, stream>>>(out, tmp, g);
  // Pass 3: conv along x (unit stride), tile y, slice z:  tmp -> out
  gauss1d_wmma<true, 1, 128, 16384><<<grid, blk, 0, stream>>>(tmp, out, g);
}